// SRFCModule_89318139887950
// MI455X (gfx1250) — compile-verified
//
#include <hip/hip_runtime.h>
#include <hip/hip_bf16.h>
#include <math.h>

#define BSZ 512
#define NP  18
#define CH  2048
#define BN_SCALE 0.9999950000374997f
#define EPSF 1e-12f

typedef __attribute__((ext_vector_type(2))) float v2f;
typedef __attribute__((ext_vector_type(8))) float v8f;

__device__ __forceinline__ v8f wmma_f32(v2f a, v2f b, v8f c) {
  // V_WMMA_F32_16X16X4_F32 : D = A(16x4) * B(4x16) + C(16x16), fp32
  return __builtin_amdgcn_wmma_f32_16x16x4_f32(false, a, false, b, (short)0, c,
                                               false, false);
}

// ---------------------------------------------------------------------------
// K1: z1 = softmax(x @ w1^T + b1), plus 1/max(||x||,eps) per (b,n) row.
// ---------------------------------------------------------------------------
__global__ void k_z1(const float* __restrict__ x, const float* __restrict__ w1,
                     const float* __restrict__ b1, float* __restrict__ z1,
                     float* __restrict__ inv) {
  const int bn = blockIdx.x;
  const int tid = threadIdx.x;
  const float* xr = x + (size_t)bn * CH;
  float s0 = 0.f, s1 = 0.f, s2 = 0.f, ss = 0.f;
  for (int c = tid; c < CH; c += 256) {
    float v = xr[c];
    ss += v * v;
    s0 += v * w1[c];
    s1 += v * w1[CH + c];
    s2 += v * w1[2 * CH + c];
  }
  __shared__ float red[4 * 256];
  red[tid] = s0; red[256 + tid] = s1; red[512 + tid] = s2; red[768 + tid] = ss;
  __syncthreads();
  for (int off = 128; off > 0; off >>= 1) {
    if (tid < off) {
      red[tid]       += red[tid + off];
      red[256 + tid] += red[256 + tid + off];
      red[512 + tid] += red[512 + tid + off];
      red[768 + tid] += red[768 + tid + off];
    }
    __syncthreads();
  }
  if (tid == 0) {
    float l0 = red[0] + b1[0], l1 = red[256] + b1[1], l2 = red[512] + b1[2];
    float m = fmaxf(l0, fmaxf(l1, l2));
    float e0 = __expf(l0 - m), e1 = __expf(l1 - m), e2 = __expf(l2 - m);
    float is = 1.f / (e0 + e1 + e2);
    z1[bn * 3 + 0] = e0 * is;
    z1[bn * 3 + 1] = e1 * is;
    z1[bn * 3 + 2] = e2 * is;
    inv[bn] = 1.f / fmaxf(sqrtf(red[768]), EPSF);
  }
}

// ---------------------------------------------------------------------------
// K2: z2 = softmax(w2b @ bn_relu(w2a @ pc + b2a) + b2b), fully fused.
// ---------------------------------------------------------------------------
__global__ void k_z2(const float* __restrict__ pc, const float* __restrict__ w2a,
                     const float* __restrict__ b2a, const float* __restrict__ g2,
                     const float* __restrict__ be2, const float* __restrict__ w2b,
                     const float* __restrict__ b2b, float* __restrict__ z2) {
  const int bn = blockIdx.x;
  const int tid = threadIdx.x;
  const float p0 = pc[bn * 4 + 0], p1 = pc[bn * 4 + 1];
  const float p2 = pc[bn * 4 + 2], p3 = pc[bn * 4 + 3];
  const float4* w4 = (const float4*)w2a;
  float t0 = 0.f, t1 = 0.f, t2 = 0.f;
  for (int c = tid; c < CH; c += 256) {
    float4 w = w4[c];
    float hv = p0 * w.x + p1 * w.y + p2 * w.z + p3 * w.w + b2a[c];
    hv = fmaxf(hv * (g2[c] * BN_SCALE) + be2[c], 0.f);
    t0 += hv * w2b[c];
    t1 += hv * w2b[CH + c];
    t2 += hv * w2b[2 * CH + c];
  }
  __shared__ float red[3 * 256];
  red[tid] = t0; red[256 + tid] = t1; red[512 + tid] = t2;
  __syncthreads();
  for (int off = 128; off > 0; off >>= 1) {
    if (tid < off) {
      red[tid]       += red[tid + off];
      red[256 + tid] += red[256 + tid + off];
      red[512 + tid] += red[512 + tid + off];
    }
    __syncthreads();
  }
  if (tid == 0) {
    float l0 = red[0] + b2b[0], l1 = red[256] + b2b[1], l2 = red[512] + b2b[2];
    float m = fmaxf(l0, fmaxf(l1, l2));
    float e0 = __expf(l0 - m), e1 = __expf(l1 - m), e2 = __expf(l2 - m);
    float is = 1.f / (e0 + e1 + e2);
    z2[bn * 3 + 0] = e0 * is;
    z2[bn * 3 + 1] = e1 * is;
    z2[bn * 3 + 2] = e2 * is;
  }
}

// ---------------------------------------------------------------------------
// K3: per-batch small outputs: d2_1, d2_2, d1_2 (18x18 each) + enc/dec.
// ---------------------------------------------------------------------------
__global__ void k_small(const float* __restrict__ z1, const float* __restrict__ z2,
                        const float* __restrict__ pc, float* __restrict__ enc,
                        float* __restrict__ dec, float* __restrict__ o_d21,
                        float* __restrict__ o_d22, float* __restrict__ o_d12) {
  const int b = blockIdx.x;
  const int tid = threadIdx.x;  // 128 threads
  __shared__ float s1n[54], s2n[54], sz[54], syc[18], sxc[18], srow[18], scol[3];
  if (tid < 54) {
    float a = z1[b * 54 + tid];
    float c = z2[b * 54 + tid];
    s1n[tid] = a; s2n[tid] = c; sz[tid] = a + c;
  }
  __syncthreads();
  if (tid < 18) {
    float a0 = s1n[tid * 3], a1 = s1n[tid * 3 + 1], a2 = s1n[tid * 3 + 2];
    float n1 = fmaxf(sqrtf(a0 * a0 + a1 * a1 + a2 * a2), EPSF);
    s1n[tid * 3] = a0 / n1; s1n[tid * 3 + 1] = a1 / n1; s1n[tid * 3 + 2] = a2 / n1;
    float c0 = s2n[tid * 3], c1 = s2n[tid * 3 + 1], c2 = s2n[tid * 3 + 2];
    float n2 = fmaxf(sqrtf(c0 * c0 + c1 * c1 + c2 * c2), EPSF);
    s2n[tid * 3] = c0 / n2; s2n[tid * 3 + 1] = c1 / n2; s2n[tid * 3 + 2] = c2 / n2;
    const float* p = pc + (size_t)b * 72 + tid * 4;
    syc[tid] = p[0] + floorf(p[2] * 0.5f);
    sxc[tid] = p[1] + floorf(p[3] * 0.5f);
    srow[tid] = sz[tid * 3] + sz[tid * 3 + 1] + sz[tid * 3 + 2];
  }
  if (tid >= 32 && tid < 35) {
    int k = tid - 32;
    float s = 0.f;
    for (int n = 0; n < 18; ++n) s += sz[n * 3 + k];
    scol[k] = s;
  }
  __syncthreads();
  if (tid < 54) {
    int n = tid / 3, k = tid % 3;
    enc[b * 54 + tid] = sz[tid] / (scol[k] + EPSF);
    dec[b * 54 + tid] = sz[tid] / (srow[n] + EPSF);
  }
  for (int idx = tid; idx < 324; idx += 128) {
    int n = idx / 18, m = idx % 18;
    float d21 = s1n[n * 3] * s1n[m * 3] + s1n[n * 3 + 1] * s1n[m * 3 + 1] +
                s1n[n * 3 + 2] * s1n[m * 3 + 2];
    float d22 = s2n[n * 3] * s2n[m * 3] + s2n[n * 3 + 1] * s2n[m * 3 + 1] +
                s2n[n * 3 + 2] * s2n[m * 3 + 2];
    float ddy = syc[m] - syc[n];
    float ddx = sxc[m] - sxc[n];
    float d12 = 1.0f - 2.0f * sqrtf(ddx * ddx + ddy * ddy);
    o_d21[b * 324 + idx] = d21;
    o_d22[b * 324 + idx] = d22;
    o_d12[b * 324 + idx] = d12;
  }
}

// ---------------------------------------------------------------------------
// K4: d1_1[b] = xn @ xn^T (18x18, K=2048) via V_WMMA_F32_16X16X4_F32.
// Out-of-range rows CLAMPED (not predicated); invalid outputs discarded at
// store. Register double-buffering overlaps loads with WMMA.
// ---------------------------------------------------------------------------
__global__ void k_d11(const float* __restrict__ x, const float* __restrict__ inv,
                      float* __restrict__ out) {
  const int b = blockIdx.x;
  const int tid = threadIdx.x;
  const int lane = tid & 31, wave = tid >> 5;
  const int tm = (wave >> 1) * 16, tn = (wave & 1) * 16;
  const int mrow = tm + (lane & 15);
  const int nrow = tn + (lane & 15);
  const int ksel = (lane >> 4) * 2;
  const float* xb = x + (size_t)b * NP * CH;
  const int mc = (mrow < NP) ? mrow : (NP - 1);
  const int nc = (nrow < NP) ? nrow : (NP - 1);
  const float* pa = xb + (size_t)mc * CH + ksel;
  const float* pb = xb + (size_t)nc * CH + ksel;

  v8f acc = {0.f, 0.f, 0.f, 0.f, 0.f, 0.f, 0.f, 0.f};
  v2f a_c = *(const v2f*)pa;
  v2f b_c = *(const v2f*)pb;
#pragma unroll 4
  for (int k = 4; k < CH; k += 4) {
    v2f a_n = *(const v2f*)(pa + k);
    v2f b_n = *(const v2f*)(pb + k);
    acc = wmma_f32(a_c, b_c, acc);
    a_c = a_n;
    b_c = b_n;
  }
  acc = wmma_f32(a_c, b_c, acc);

  const int nOut = tn + (lane & 15);
#pragma unroll
  for (int r = 0; r < 8; ++r) {
    int mOut = tm + r + 8 * (lane >> 4);
    if (mOut < NP && nOut < NP)
      out[b * 324 + mOut * 18 + nOut] =
          acc[r] * inv[b * NP + mOut] * inv[b * NP + nOut];
  }
}

// ---------------------------------------------------------------------------
// K5: u[b,k,c] = sum_n enc[b,n,k] * x[b,n,c]
// ---------------------------------------------------------------------------
__global__ void k_u(const float* __restrict__ x, const float* __restrict__ enc,
                    float* __restrict__ u) {
  const int b = blockIdx.y;
  const int c = blockIdx.x * 256 + threadIdx.x;
  __shared__ float se[54];
  if (threadIdx.x < 54) se[threadIdx.x] = enc[b * 54 + threadIdx.x];
  __syncthreads();
  float a0 = 0.f, a1 = 0.f, a2 = 0.f;
  const float* xb = x + (size_t)b * NP * CH + c;
#pragma unroll
  for (int n = 0; n < NP; ++n) {
    float xv = xb[(size_t)n * CH];
    a0 += se[n * 3 + 0] * xv;
    a1 += se[n * 3 + 1] * xv;
    a2 += se[n * 3 + 2] * xv;
  }
  float* ub = u + (size_t)b * 3 * CH + c;
  ub[0] = a0; ub[CH] = a1; ub[2 * CH] = a2;
}

// ---------------------------------------------------------------------------
// K6/K8: OUT[m,d] = relu((A[m,:]·W[d,:] + bias[d]) * (gamma[d]*BN_SCALE)
//                        + beta[d]) (+ resid[m,d])
// Wave tile 32x64: 2 A-fragments x 4 B-fragments -> 8 accumulators,
// 6 b64 loads per 8 WMMAs (0.75 loads/WMMA, 10.7 flops/byte from L2).
// Block = 8 waves along N -> 32x512 tile. Register double-buffered K loop.
// ---------------------------------------------------------------------------
__global__ void k_gemm_bnrelu(const float* __restrict__ A, const float* __restrict__ W,
                              const float* __restrict__ bias, const float* __restrict__ gamma,
                              const float* __restrict__ beta, const float* __restrict__ resid,
                              float* __restrict__ out) {
  const int tid = threadIdx.x;
  const int lane = tid & 31, wave = tid >> 5;
  const int mBase = blockIdx.x * 32;
  const int nBase = blockIdx.y * 512 + wave * 64;
  const int mrow0 = mBase + (lane & 15);
  const int ncol = nBase + (lane & 15);
  const int ksel = (lane >> 4) * 2;
  const float* Ar0 = A + (size_t)mrow0 * CH + ksel;
  const float* Ar1 = Ar0 + (size_t)16 * CH;
  const float* B0 = W + (size_t)ncol * CH + ksel;

  v8f acc00 = {0.f, 0.f, 0.f, 0.f, 0.f, 0.f, 0.f, 0.f};
  v8f acc01 = acc00, acc02 = acc00, acc03 = acc00;
  v8f acc10 = acc00, acc11 = acc00, acc12 = acc00, acc13 = acc00;

  v2f a0_c = *(const v2f*)(Ar0);
  v2f a1_c = *(const v2f*)(Ar1);
  v2f b0_c = *(const v2f*)(B0);
  v2f b1_c = *(const v2f*)(B0 + 16 * CH);
  v2f b2_c = *(const v2f*)(B0 + 32 * CH);
  v2f b3_c = *(const v2f*)(B0 + 48 * CH);
#pragma unroll 2
  for (int k = 4; k < CH; k += 4) {
    v2f a0_n = *(const v2f*)(Ar0 + k);
    v2f a1_n = *(const v2f*)(Ar1 + k);
    v2f b0_n = *(const v2f*)(B0 + k);
    v2f b1_n = *(const v2f*)(B0 + 16 * CH + k);
    v2f b2_n = *(const v2f*)(B0 + 32 * CH + k);
    v2f b3_n = *(const v2f*)(B0 + 48 * CH + k);
    acc00 = wmma_f32(a0_c, b0_c, acc00);
    acc01 = wmma_f32(a0_c, b1_c, acc01);
    acc02 = wmma_f32(a0_c, b2_c, acc02);
    acc03 = wmma_f32(a0_c, b3_c, acc03);
    acc10 = wmma_f32(a1_c, b0_c, acc10);
    acc11 = wmma_f32(a1_c, b1_c, acc11);
    acc12 = wmma_f32(a1_c, b2_c, acc12);
    acc13 = wmma_f32(a1_c, b3_c, acc13);
    a0_c = a0_n; a1_c = a1_n;
    b0_c = b0_n; b1_c = b1_n; b2_c = b2_n; b3_c = b3_n;
  }
  acc00 = wmma_f32(a0_c, b0_c, acc00);
  acc01 = wmma_f32(a0_c, b1_c, acc01);
  acc02 = wmma_f32(a0_c, b2_c, acc02);
  acc03 = wmma_f32(a0_c, b3_c, acc03);
  acc10 = wmma_f32(a1_c, b0_c, acc10);
  acc11 = wmma_f32(a1_c, b1_c, acc11);
  acc12 = wmma_f32(a1_c, b2_c, acc12);
  acc13 = wmma_f32(a1_c, b3_c, acc13);

  v8f accs[2][4] = {{acc00, acc01, acc02, acc03}, {acc10, acc11, acc12, acc13}};
#pragma unroll
  for (int t = 0; t < 4; ++t) {
    const int Nd = nBase + t * 16 + (lane & 15);
    const float bv = bias[Nd];
    const float gg = gamma[Nd] * BN_SCALE;
    const float bb = beta[Nd];
#pragma unroll
    for (int m = 0; m < 2; ++m) {
#pragma unroll
      for (int r = 0; r < 8; ++r) {
        const int M = mBase + m * 16 + r + 8 * (lane >> 4);
        float v = accs[m][t][r] + bv;
        v = fmaxf(v * gg + bb, 0.f);
        if (resid) v += resid[(size_t)M * CH + Nd];
        out[(size_t)M * CH + Nd] = v;
      }
    }
  }
}

// ---------------------------------------------------------------------------
// K7: y_pre[b,n,c] = sum_k dec[b,n,k] * uc[b,k,c]
// ---------------------------------------------------------------------------
__global__ void k_y(const float* __restrict__ uc, const float* __restrict__ dec,
                    float* __restrict__ ypre) {
  const int b = blockIdx.y;
  const int c = blockIdx.x * 256 + threadIdx.x;
  __shared__ float sd[54];
  if (threadIdx.x < 54) sd[threadIdx.x] = dec[b * 54 + threadIdx.x];
  __syncthreads();
  const float* ub = uc + (size_t)b * 3 * CH + c;
  const float u0 = ub[0], u1 = ub[CH], u2 = ub[2 * CH];
  float* yb = ypre + (size_t)b * NP * CH + c;
#pragma unroll
  for (int n = 0; n < NP; ++n)
    yb[(size_t)n * CH] = sd[n * 3] * u0 + sd[n * 3 + 1] * u1 + sd[n * 3 + 2] * u2;
}

// ---------------------------------------------------------------------------
extern "C" void kernel_launch(void* const* d_in, const int* in_sizes, int n_in,
                              void* d_out, int out_size, void* d_ws, size_t ws_size,
                              hipStream_t stream) {
  (void)in_sizes; (void)n_in; (void)out_size; (void)ws_size;
  const float* x   = (const float*)d_in[0];
  const float* pc  = (const float*)d_in[1];
  const float* w1  = (const float*)d_in[2];
  const float* b1  = (const float*)d_in[3];
  const float* w2a = (const float*)d_in[4];
  const float* b2a = (const float*)d_in[5];
  const float* g2  = (const float*)d_in[6];
  const float* be2 = (const float*)d_in[7];
  const float* w2b = (const float*)d_in[8];
  const float* b2b = (const float*)d_in[9];
  const float* wc  = (const float*)d_in[10];
  const float* bc  = (const float*)d_in[11];
  const float* gc  = (const float*)d_in[12];
  const float* bec = (const float*)d_in[13];
  const float* w3  = (const float*)d_in[14];
  const float* b3  = (const float*)d_in[15];
  const float* g3  = (const float*)d_in[16];
  const float* be3 = (const float*)d_in[17];

  float* out = (float*)d_out;
  const size_t Y_LEN  = (size_t)BSZ * NP * CH;   // 18874368
  const size_t D_LEN  = (size_t)BSZ * NP * NP;   // 165888
  float* o_y   = out;
  float* o_d21 = out + Y_LEN;
  float* o_d11 = out + Y_LEN + D_LEN;
  float* o_d22 = out + Y_LEN + 2 * D_LEN;
  float* o_d12 = out + Y_LEN + 3 * D_LEN;

  float* ws = (float*)d_ws;
  float* z1   = ws;                              // 27648
  float* z2   = z1 + (size_t)BSZ * NP * 3;       // 27648
  float* inv  = z2 + (size_t)BSZ * NP * 3;       // 9216
  float* enc  = inv + (size_t)BSZ * NP;          // 27648
  float* dec  = enc + (size_t)BSZ * NP * 3;      // 27648
  float* u    = dec + (size_t)BSZ * NP * 3;      // 3145728
  float* uc   = u + (size_t)BSZ * 3 * CH;        // 3145728
  float* ypre = uc + (size_t)BSZ * 3 * CH;       // 18874368

  const int BN = BSZ * NP;  // 9216

  k_z1<<<BN, 256, 0, stream>>>(x, w1, b1, z1, inv);
  k_z2<<<BN, 256, 0, stream>>>(pc, w2a, b2a, g2, be2, w2b, b2b, z2);
  k_small<<<BSZ, 128, 0, stream>>>(z1, z2, pc, enc, dec, o_d21, o_d22, o_d12);
  k_d11<<<BSZ, 128, 0, stream>>>(x, inv, o_d11);
  k_u<<<dim3(CH / 256, BSZ), 256, 0, stream>>>(x, enc, u);
  // uc = bn_relu(u @ wc^T + bc): M = 1536 -> 48 m-blocks, 4 n-blocks
  k_gemm_bnrelu<<<dim3(48, 4), 256, 0, stream>>>(u, wc, bc, gc, bec, nullptr, uc);
  k_y<<<dim3(CH / 256, BSZ), 256, 0, stream>>>(uc, dec, ypre);
  // y = bn_relu(ypre @ w3^T + b3) + x: M = 9216 -> 288 m-blocks, 4 n-blocks
  k_gemm_bnrelu<<<dim3(288, 4), 256, 0, stream>>>(ypre, w3, b3, g3, be3, x, o_y);
}